// GGNNMsgPass_31834297598504
// MI455X (gfx1250) — compile-verified
//
#include <hip/hip_runtime.h>
#include <hip/hip_bf16.h>

// GGNN message passing for MI455X (gfx1250, wave32).
// B=16, N=512, D=64, K=8.
//
// Phase 1 (WMMA f32 16x16x4): T[b,j,k,d] = sum_e M[k,d,e] * h[b,j,e]
// Phase 2: adjacency transpose (LDS tiled) for the "out" direction
// Phase 3: gather-accumulate over j with edge-class select (float4 / b128
//          loads, L2-resident T working set), + bias.

#define Bsz 16
#define Nn  512
#define Dd  64
#define Kk  8
#define BN  (Bsz * Nn)             // 8192 rows
#define TNF ((size_t)BN * Kk * Dd) // floats per T buffer = 4,194,304

typedef __attribute__((ext_vector_type(2))) float v2f;
typedef __attribute__((ext_vector_type(4))) float v4f;
typedef __attribute__((ext_vector_type(8))) float v8f;

// ---------------------------------------------------------------------------
// Kernel 1: transform. One wave per job = (dir, k, j-tile of 16 rows).
// jobs = 2 * 8 * 512 = 8192 waves -> 1024 blocks x 256 threads (8 waves).
// A (16x4, f32): lanes 0-15 hold M=row, VGPR0 = K {0|2}, VGPR1 = K {1|3}.
// B (4x16, f32): mirror layout, N = lane&15, rows K in half-select.
// C (16x16 f32): VGPR r -> M = r + 8*(lane>=16), N = lane&15.
// ---------------------------------------------------------------------------
__global__ void __launch_bounds__(256)
ggnn_transform_kernel(const float* __restrict__ h,
                      const float* __restrict__ Min,
                      const float* __restrict__ Mout,
                      float* __restrict__ T_in,
                      float* __restrict__ T_out) {
    const int wid  = (int)((blockIdx.x * blockDim.x + threadIdx.x) >> 5);
    const int lane = (int)(threadIdx.x & 31);

    const int dir  = wid >> 12;        // 0 = in, 1 = out
    const int rem  = wid & 4095;
    const int k    = rem >> 9;         // edge class index 0..7 (class k+1)
    const int jt   = rem & 511;        // j-tile over flattened B*N
    const int row0 = jt << 4;

    const int halfsel = lane >> 4;     // 0: lanes 0-15, 1: lanes 16-31
    const int lrow    = lane & 15;

    // Load the full 16x64 A tile once (reused for all 4 d-tiles).
    v2f a[16];
    const float* hrow = h + (size_t)(row0 + lrow) * Dd;
#pragma unroll
    for (int ec = 0; ec < 16; ++ec) {
        const int e = ec * 4 + halfsel * 2;
        a[ec].x = hrow[e];
        a[ec].y = hrow[e + 1];
    }

    const float* M = (dir ? Mout : Min) + (size_t)k * Dd * Dd; // M[k][d][e]
    float* T       = dir ? T_out : T_in;

#pragma unroll
    for (int dt = 0; dt < 4; ++dt) {
        const int d = dt * 16 + lrow;  // output column for this lane
        v8f c = {};
#pragma unroll
        for (int ec = 0; ec < 16; ++ec) {
            const int e = ec * 4 + halfsel * 2;
            v2f bv;
            bv.x = M[(size_t)d * Dd + e];       // B[e][d] = M[d][e]
            bv.y = M[(size_t)d * Dd + e + 1];
            c = __builtin_amdgcn_wmma_f32_16x16x4_f32(
                    /*neg_a=*/false, a[ec],
                    /*neg_b=*/false, bv,
                    /*c_mod=*/(short)0, c,
                    /*reuse_a=*/false, /*reuse_b=*/false);
        }
#pragma unroll
        for (int r = 0; r < 8; ++r) {
            const int row = row0 + r + halfsel * 8;
            T[((size_t)row * Kk + k) * Dd + d] = c[r];
        }
    }
}

// ---------------------------------------------------------------------------
// Kernel 2: adjacency transpose (per batch), 16x16 LDS tiles.
// adjT[b][i][j] = adj[b][j][i]
// ---------------------------------------------------------------------------
__global__ void __launch_bounds__(256)
ggnn_transpose_kernel(const int* __restrict__ adj, int* __restrict__ adjT) {
    __shared__ int tile[16][17];
    const int b  = (int)blockIdx.z;
    const int i0 = (int)blockIdx.y * 16;
    const int j0 = (int)blockIdx.x * 16;
    const int tx = (int)threadIdx.x;
    const int ty = (int)threadIdx.y;

    const int* base = adj + (size_t)b * Nn * Nn;
    tile[ty][tx] = base[(size_t)(i0 + ty) * Nn + (j0 + tx)];
    __syncthreads();
    int* baseT = adjT + (size_t)b * Nn * Nn;
    baseT[(size_t)(j0 + ty) * Nn + (i0 + tx)] = tile[tx][ty];
}

// ---------------------------------------------------------------------------
// Kernel 3: gather-accumulate. One block (256 threads) per (b, i) row.
// Threads = 16 j-lanes x 16 d-quarters; each thread owns a float4 of the
// feature vector -> global_load_b128 per (edge, thread), 256B/row coalesced.
// Edge classes staged through LDS (broadcast reads, conflict-free).
// Per-batch T working set is 2 MB -> lives in the 192 MB L2 across re-reads.
// ---------------------------------------------------------------------------
__global__ void __launch_bounds__(256)
ggnn_gather_kernel(const int* __restrict__ adjIn,
                   const int* __restrict__ adjT,
                   const float* __restrict__ T_in,
                   const float* __restrict__ T_out,
                   const float* __restrict__ bias,
                   float* __restrict__ out) {
    const int bi  = (int)blockIdx.x;      // b*N + i
    const int tid = (int)threadIdx.x;
    const int g   = tid >> 4;             // j-lane 0..15
    const int dq  = tid & 15;             // d-quarter: d = dq*4 .. dq*4+3
    const int b   = bi >> 9;

    __shared__ int kin_s[256];
    __shared__ int kout_s[256];
    __shared__ v4f red[2][16][16];

    const int* rowIn  = adjIn + (size_t)bi * Nn;
    const int* rowOut = adjT  + (size_t)bi * Nn;
    const float* Tib  = T_in  + (size_t)b * Nn * Kk * Dd + dq * 4;
    const float* Tob  = T_out + (size_t)b * Nn * Kk * Dd + dq * 4;

    v4f accI = {0.0f, 0.0f, 0.0f, 0.0f};
    v4f accO = {0.0f, 0.0f, 0.0f, 0.0f};

    for (int jc = 0; jc < Nn; jc += 256) {
        // prefetch next chunk of edge classes while we work on this one
        if (jc + 256 < Nn) {
            __builtin_prefetch(&rowIn[jc + 256 + tid], 0, 1);
            __builtin_prefetch(&rowOut[jc + 256 + tid], 0, 1);
        }
        kin_s[tid]  = rowIn[jc + tid];
        kout_s[tid] = rowOut[jc + tid];
        __syncthreads();

        for (int jj = g; jj < 256; jj += 16) {
            const int j  = jc + jj;
            const int ki = kin_s[jj];
            if (ki > 0) {
                const v4f v = *(const v4f*)&Tib[((size_t)j * Kk + (ki - 1)) * Dd];
                accI += v;
            }
            const int ko = kout_s[jj];
            if (ko > 0) {
                const v4f v = *(const v4f*)&Tob[((size_t)j * Kk + (ko - 1)) * Dd];
                accO += v;
            }
        }
        __syncthreads();
    }

    red[0][g][dq] = accI;
    red[1][g][dq] = accO;
    __syncthreads();

    if (g == 0) {
        v4f si = red[0][0][dq];
        v4f so = red[1][0][dq];
#pragma unroll
        for (int t = 1; t < 16; ++t) {
            si += red[0][t][dq];
            so += red[1][t][dq];
        }
        const v4f bi_in  = *(const v4f*)&bias[dq * 4];
        const v4f bi_out = *(const v4f*)&bias[Dd + dq * 4];
        si += bi_in;
        so += bi_out;
        *(v4f*)&out[(size_t)bi * (2 * Dd) + dq * 4]      = si;
        *(v4f*)&out[(size_t)bi * (2 * Dd) + Dd + dq * 4] = so;
    }
}

// ---------------------------------------------------------------------------
// Launch
// ---------------------------------------------------------------------------
extern "C" void kernel_launch(void* const* d_in, const int* in_sizes, int n_in,
                              void* d_out, int out_size, void* d_ws, size_t ws_size,
                              hipStream_t stream) {
    const float* node_states = (const float*)d_in[0];   // [B,N,D]
    const int*   adjacency   = (const int*)d_in[1];     // [B,N,N]
    // d_in[2] = distance, unused (as in reference)
    const float* matrices_in  = (const float*)d_in[3];  // [K,D,D]
    const float* matrices_out = (const float*)d_in[4];  // [K,D,D]
    const float* message_bias = (const float*)d_in[5];  // [2D]
    float* out = (float*)d_out;                         // [B,N,2D]

    float* T_in  = (float*)d_ws;
    float* T_out = T_in + TNF;
    int*   adjT  = (int*)(T_out + TNF);

    // Phase 1: 8192 waves (2 dir x 8 k x 512 j-tiles) -> 1024 blocks of 256.
    ggnn_transform_kernel<<<1024, 256, 0, stream>>>(
        node_states, matrices_in, matrices_out, T_in, T_out);

    // Phase 2: adjacency transpose.
    dim3 tgrid(Nn / 16, Nn / 16, Bsz);
    dim3 tblock(16, 16, 1);
    ggnn_transpose_kernel<<<tgrid, tblock, 0, stream>>>(adjacency, adjT);

    // Phase 3: gather. One block per (b, i).
    ggnn_gather_kernel<<<BN, 256, 0, stream>>>(
        adjacency, adjT, T_in, T_out, message_bias, out);
}